// S3Ternary_44212393345428
// MI455X (gfx1250) — compile-verified
//
#include <hip/hip_runtime.h>
#include <hip/hip_bf16.h>
#include <math.h>

typedef float v2f __attribute__((ext_vector_type(2)));
typedef float v8f __attribute__((ext_vector_type(8)));

// ---- problem constants (from setup_inputs) ----
// delta: (4, 8192, 2048) f32 ; d_model=2048, d_align=512, reg_real=1536
// pa_w (3,512,1536) pd_w (3,512,2048) wr_w/wi_w (9,256,2048) wg_w (9,1,2048)
#define D_MODEL   2048
#define N_ROWS    32768            // B*L = 4*8192
#define GATED_N   67108864ull      // 4*8192*2048

// ---- workspace layout (float indices into d_ws) ----
#define WS_PARTIAL 0               // 64 * 2048 row-chunk partial sums
#define WS_ABSPART 131072          // 76 abs-sum partials (pa:12, pd:16, wr:3*8, wi:3*8)
#define WS_SCALES  131148          // 8 : [s_pa, s_pd, s_wr0..2, s_wi0..2]
#define WS_SUMMARY 131156          // 2048
#define WS_DDIR    133204          // 2048
#define WS_RDIR    135252          // 1536
#define WS_DPROJ   136788          // 512
#define WS_RPROJ   137300          // 512
#define WS_UPD     137812          // 1536 : upd_r (3*256) then upd_i (3*256)
#define WS_WG      139348          // 3
#define WS_GATE    139351          // 1

// ---- output layout (floats) ----
#define OUT_REG_OFF  GATED_N            // updated_registers: 1536
#define OUT_GATE_OFF (GATED_N + 1536)   // gate: 1
#define OUT_WG_OFF   (GATED_N + 1537)   // wg: 3

// =====================================================================
// Kernel 1: column partial-sums of delta via WMMA ones-GEMM.
// grid (16, 64), block 256 (8 waves). Wave w of block bx owns 16 columns
// c0 = bx*128 + w*16; block by owns rows [by*512, by*512+512).
// A(16x4) = all ones, B(4x16) = delta tile -> D[m,n] = sum_k delta[k,n].
// B f32 layout: VGPR0 lanes0-15 K=0, lanes16-31 K=2; VGPR1 K=1 / K=3.
// =====================================================================
__global__ void k_sum_partial(const float* __restrict__ delta,
                              float* __restrict__ partial) {
    const int lane = threadIdx.x & 31;
    const int wave = threadIdx.x >> 5;
    const int c0   = blockIdx.x * 128 + wave * 16;
    const int colA = c0 + (lane & 15);
    const int rb   = (lane >> 4) * 2;            // 0 for lanes 0-15, 2 for 16-31
    const float* p = delta + (size_t)(blockIdx.y * 512 + rb) * D_MODEL + colA;

    v2f a; a.x = 1.0f; a.y = 1.0f;               // ones A-matrix
    v8f acc0 = {}; v8f acc1 = {};
    for (int r = 0; r < 512; r += 8) {
        v2f b0; b0.x = p[0];            b0.y = p[D_MODEL];
        v2f b1; b1.x = p[4 * D_MODEL];  b1.y = p[5 * D_MODEL];
        p += 8 * D_MODEL;
        acc0 = __builtin_amdgcn_wmma_f32_16x16x4_f32(
            false, a, false, b0, (short)0, acc0, false, false);
        acc1 = __builtin_amdgcn_wmma_f32_16x16x4_f32(
            false, a, false, b1, (short)0, acc1, false, false);
    }
    v8f acc = acc0 + acc1;
    // row M=0 of D (column sums) lives in element 0 for lanes 0-15
    if (lane < 16) partial[blockIdx.y * D_MODEL + colA] = acc[0];
}

// =====================================================================
// Kernel 2: per-matrix |w| partial sums for ternary absmean scales.
// Matrix selected on-device: offset = (phase*matPerPhase + matIdx)*matElems.
// =====================================================================
__global__ void k_absmean_partial(const float* __restrict__ base,
                                  const int* __restrict__ phase,
                                  int matPerPhase, int matIdx, long long matElems,
                                  float* __restrict__ out, int chunk) {
    const float* src = base + ((long long)(*phase) * matPerPhase + matIdx) * matElems;
    long long start = (long long)blockIdx.x * chunk;
    long long end   = start + chunk; if (end > matElems) end = matElems;
    float s = 0.0f;
    for (long long i = start + threadIdx.x; i < end; i += blockDim.x)
        s += fabsf(src[i]);
    __shared__ float red[256];
    red[threadIdx.x] = s; __syncthreads();
    for (int o = 128; o > 0; o >>= 1) {
        if (threadIdx.x < o) red[threadIdx.x] += red[threadIdx.x + o];
        __syncthreads();
    }
    if (threadIdx.x == 0) out[blockIdx.x] = red[0];
}

// =====================================================================
// Kernel 3 (1 block): scales, summary, delta_dir, reg_dir, wg.
// =====================================================================
__global__ void k_finalize1(const float* __restrict__ registers_in,
                            const float* __restrict__ wg_w,
                            const float* __restrict__ wg_b,
                            const int* __restrict__ phase,
                            float* __restrict__ ws, float* __restrict__ out) {
    __shared__ float red[256];
    const int tid = threadIdx.x;
    const int ph  = *phase;

    // ternary scales: s = mean|w| + EPS
    if (tid < 8) {
        const int starts[9] = {0, 12, 28, 36, 44, 52, 60, 68, 76};
        const float elems[8] = {786432.f, 1048576.f, 524288.f, 524288.f,
                                524288.f, 524288.f, 524288.f, 524288.f};
        float ss = 0.0f;
        for (int i = starts[tid]; i < starts[tid + 1]; ++i) ss += ws[WS_ABSPART + i];
        ws[WS_SCALES + tid] = ss / elems[tid] + 1e-8f;
    }

    // summary = mean over rows; accumulate |summary|^2
    float sq = 0.0f;
    for (int col = tid; col < D_MODEL; col += 256) {
        float s = 0.0f;
        for (int c = 0; c < 64; ++c) s += ws[WS_PARTIAL + c * D_MODEL + col];
        s *= (1.0f / (float)N_ROWS);
        ws[WS_SUMMARY + col] = s;
        sq += s * s;
    }
    red[tid] = sq; __syncthreads();
    for (int o = 128; o > 0; o >>= 1) {
        if (tid < o) red[tid] += red[tid + o];
        __syncthreads();
    }
    const float dn = sqrtf(red[0]) + 1e-8f;
    __syncthreads();
    for (int col = tid; col < D_MODEL; col += 256)
        ws[WS_DDIR + col] = ws[WS_SUMMARY + col] / dn;

    // reg_dir
    float rq = 0.0f;
    for (int i = tid; i < 1536; i += 256) { float v = registers_in[i]; rq += v * v; }
    red[tid] = rq; __syncthreads();
    for (int o = 128; o > 0; o >>= 1) {
        if (tid < o) red[tid] += red[tid + o];
        __syncthreads();
    }
    const float rn = sqrtf(red[0]) + 1e-8f;
    __syncthreads();
    for (int i = tid; i < 1536; i += 256)
        ws[WS_RDIR + i] = registers_in[i] / rn;

    // wg = sigmoid(wg_w[sl] . summary + wg_b[sl])
    for (int r = 0; r < 3; ++r) {
        const float* w = wg_w + (size_t)(ph * 3 + r) * D_MODEL;
        float a = 0.0f;
        for (int k = tid; k < D_MODEL; k += 256) a += w[k] * ws[WS_SUMMARY + k];
        red[tid] = a; __syncthreads();
        for (int o = 128; o > 0; o >>= 1) {
            if (tid < o) red[tid] += red[tid + o];
            __syncthreads();
        }
        if (tid == 0) {
            float g = 1.0f / (1.0f + expf(-(red[0] + wg_b[ph * 3 + r])));
            ws[WS_WG + r]     = g;
            out[OUT_WG_OFF + r] = g;
        }
        __syncthreads();
    }
}

// =====================================================================
// Kernel 4: all ternary matvec rows. One block per output row.
//   rows    0..511  : delta_proj = Tq(pd_w[ph]) . delta_dir
//   rows  512..1023 : reg_proj   = Tq(pa_w[ph]) . reg_dir
//   rows 1024..2559 : upd_r/upd_i rows = Tq(wr/wi[sl]) . summary
// quantize q = clip(rint(w/s), -1, 1); dot scaled by s at the end.
// =====================================================================
__global__ void k_proj(const float* __restrict__ pa_w, const float* __restrict__ pd_w,
                       const float* __restrict__ wr_w, const float* __restrict__ wi_w,
                       const int* __restrict__ phase, float* __restrict__ ws) {
    __shared__ float red[256];
    const int tid = threadIdx.x, row = blockIdx.x, ph = *phase;
    const float* W; const float* x; int K; float s; int outIdx;
    if (row < 512) {
        W = pd_w + (size_t)ph * 512 * 2048 + (size_t)row * 2048;
        x = ws + WS_DDIR; K = 2048; s = ws[WS_SCALES + 1]; outIdx = WS_DPROJ + row;
    } else if (row < 1024) {
        int r = row - 512;
        W = pa_w + (size_t)ph * 512 * 1536 + (size_t)r * 1536;
        x = ws + WS_RDIR; K = 1536; s = ws[WS_SCALES + 0]; outIdx = WS_RPROJ + r;
    } else {
        int t  = row - 1024;      // 0..1535
        int m6 = t >> 8;          // 0..2 -> wr, 3..5 -> wi
        int rk = t & 255;
        const float* base = (m6 < 3) ? wr_w : wi_w;
        int m = (m6 < 3) ? m6 : m6 - 3;
        W = base + (size_t)(ph * 3 + m) * 524288 + (size_t)rk * 2048;
        x = ws + WS_SUMMARY; K = 2048; s = ws[WS_SCALES + 2 + m6]; outIdx = WS_UPD + t;
    }
    const float inv_s = 1.0f / s;
    float acc = 0.0f;
    for (int k = tid; k < K; k += 256) {
        float q = rintf(W[k] * inv_s);
        q = fminf(1.0f, fmaxf(-1.0f, q));
        acc += q * x[k];
    }
    red[tid] = acc; __syncthreads();
    for (int o = 128; o > 0; o >>= 1) {
        if (tid < o) red[tid] += red[tid + o];
        __syncthreads();
    }
    if (tid == 0) ws[outIdx] = red[0] * s;
}

// =====================================================================
// Kernel 5 (1 block): alignment, gate, updated_registers.
// =====================================================================
__global__ void k_finalize2(const float* __restrict__ registers_in,
                            const float* __restrict__ temperature,
                            const float* __restrict__ learned_bias,
                            const int* __restrict__ phase,
                            float* __restrict__ ws, float* __restrict__ out) {
    __shared__ float red[256];
    const int tid = threadIdx.x;
    float a = 0.0f;
    for (int i = tid; i < 512; i += 256)
        a += ws[WS_DPROJ + i] * ws[WS_RPROJ + i];
    red[tid] = a; __syncthreads();
    for (int o = 128; o > 0; o >>= 1) {
        if (tid < o) red[tid] += red[tid + o];
        __syncthreads();
    }
    const int ph = *phase;
    if (tid == 0) {
        float g = 1.0f / (1.0f + expf(-(red[0] * temperature[ph] + learned_bias[ph])));
        ws[WS_GATE]       = g;
        out[OUT_GATE_OFF] = g;
    }
    __syncthreads();
    // updated_registers = registers + wg[r] * upd ; layout (r, k, [re,im])
    for (int i = tid; i < 1536; i += 256) {
        int r = i / 512, k = (i % 512) >> 1, comp = i & 1;
        float u = ws[WS_UPD + comp * 768 + r * 256 + k];
        out[OUT_REG_OFF + i] = registers_in[i] + ws[WS_WG + r] * u;
    }
}

// =====================================================================
// Kernel 6: gated_delta = gate * delta ; float4 streaming pass (536 MB).
// =====================================================================
__global__ void k_gate(const float* __restrict__ delta,
                       const float* __restrict__ ws, float* __restrict__ out) {
    const float g = ws[WS_GATE];
    const float4* in4 = (const float4*)delta;
    float4* out4 = (float4*)out;
    size_t i      = (size_t)blockIdx.x * blockDim.x + threadIdx.x;
    size_t stride = (size_t)gridDim.x * blockDim.x;
    for (size_t j = i; j < (GATED_N / 4); j += stride) {
        float4 v = in4[j];
        v.x *= g; v.y *= g; v.z *= g; v.w *= g;
        out4[j] = v;
    }
}

extern "C" void kernel_launch(void* const* d_in, const int* in_sizes, int n_in,
                              void* d_out, int out_size, void* d_ws, size_t ws_size,
                              hipStream_t stream) {
    const float* delta       = (const float*)d_in[0];
    const float* registers   = (const float*)d_in[1];
    const float* pa_w        = (const float*)d_in[2];
    const float* pd_w        = (const float*)d_in[3];
    const float* temperature = (const float*)d_in[4];
    const float* learned_bias= (const float*)d_in[5];
    const float* wr_w        = (const float*)d_in[6];
    const float* wi_w        = (const float*)d_in[7];
    const float* wg_w        = (const float*)d_in[8];
    const float* wg_b        = (const float*)d_in[9];
    const int*   phase       = (const int*)d_in[10];
    float* out = (float*)d_out;
    float* ws  = (float*)d_ws;

    // 1) column partial sums of delta (WMMA ones-GEMM)
    k_sum_partial<<<dim3(16, 64), 256, 0, stream>>>(delta, ws + WS_PARTIAL);

    // 2) absmean partials for the 8 ternary matrices of this phase
    k_absmean_partial<<<12, 256, 0, stream>>>(pa_w, phase, 1, 0, 786432LL,
                                              ws + WS_ABSPART + 0, 65536);
    k_absmean_partial<<<16, 256, 0, stream>>>(pd_w, phase, 1, 0, 1048576LL,
                                              ws + WS_ABSPART + 12, 65536);
    for (int m = 0; m < 3; ++m)
        k_absmean_partial<<<8, 256, 0, stream>>>(wr_w, phase, 3, m, 524288LL,
                                                 ws + WS_ABSPART + 28 + 8 * m, 65536);
    for (int m = 0; m < 3; ++m)
        k_absmean_partial<<<8, 256, 0, stream>>>(wi_w, phase, 3, m, 524288LL,
                                                 ws + WS_ABSPART + 52 + 8 * m, 65536);

    // 3) summary / directions / scales / wg
    k_finalize1<<<1, 256, 0, stream>>>(registers, wg_w, wg_b, phase, ws, out);

    // 4) all ternary matvec rows
    k_proj<<<2560, 256, 0, stream>>>(pa_w, pd_w, wr_w, wi_w, phase, ws);

    // 5) gate + updated registers
    k_finalize2<<<1, 256, 0, stream>>>(registers, temperature, learned_bias, phase, ws, out);

    // 6) streaming gated_delta
    k_gate<<<8192, 256, 0, stream>>>(delta, ws, out);
}